// StableComplexRBFKAN_23957327577257
// MI455X (gfx1250) — compile-verified
//
#include <hip/hip_runtime.h>
#include <math.h>

typedef float v2f __attribute__((ext_vector_type(2)));
typedef float v8f __attribute__((ext_vector_type(8)));

#define IN_DIM 9
#define HID    16
#define NRBF   32
#define NRBF2  16

// workspace layout (float offsets)
#define OFF_B1   0      // [4 tiles][6 chunks][32 lanes][2]
#define OFF_B2   1536   // [2][8][32][2]
#define OFF_B3   2560   // [2][8][32][2]
#define OFF_B4   3584   // [4][32][2]
#define OFF_G1   3840   // [2][32]
#define OFF_CS1  3904   // [2][32]
#define OFF_BIA  3968   // [2][32]
#define OFF_G2   4032   // [32]
#define OFF_CS2  4064   // [32]
#define OFF_BOUT 4096   // [32]

// ---------------------------------------------------------------------------
// Pack kernel: rewrite tiny weights into WMMA B-layout (K padded: re 0..8 pad
// 9..11, im 12..20 pad 21..23). One wave, lane L owns B[*][*][L][*].
// ---------------------------------------------------------------------------
__global__ void pack_kernel(
    const float* __restrict__ w1_re, const float* __restrict__ w1_im,
    const float* __restrict__ b1_re, const float* __restrict__ b1_im,
    const float* __restrict__ c1_re, const float* __restrict__ c1_im,
    const float* __restrict__ log_g1,
    const float* __restrict__ rw1_re, const float* __restrict__ rw1_im,
    const float* __restrict__ w2_re, const float* __restrict__ w2_im,
    const float* __restrict__ b2_re, const float* __restrict__ b2_im,
    const float* __restrict__ c2_re, const float* __restrict__ c2_im,
    const float* __restrict__ log_g2,
    const float* __restrict__ rw2_re, const float* __restrict__ rw2_im,
    const float* __restrict__ scale1, const float* __restrict__ scale2,
    float* __restrict__ ws)
{
    const int L = threadIdx.x & 31;
    const int half = L >> 4, r = L & 15;
    const float s1 = scale1[0], s2 = scale2[0];

    // ---- B1: stage-1 GEMM  A=[x_re | x_im] (K=24 padded), 4 output tiles
    for (int t = 0; t < 4; ++t)
        for (int c = 0; c < 6; ++c)
            for (int j = 0; j < 2; ++j) {
                int K = 4 * c + 2 * half + j;
                float v = 0.f;
                if (t == 0) {            // h_re columns
                    if (K < 9)                    v =  w1_re[r * 9 + K];
                    else if (K >= 12 && K < 21)   v = -w1_im[r * 9 + (K - 12)];
                } else if (t == 1) {     // h_im columns
                    if (K < 9)                    v =  w1_im[r * 9 + K];
                    else if (K >= 12 && K < 21)   v =  w1_re[r * 9 + (K - 12)];
                } else {                 // cross Re(x conj(c1)) columns
                    int kc = r + 16 * (t - 2);
                    if (K < 9)                    v =  c1_re[kc * 9 + K];
                    else if (K >= 12 && K < 21)   v =  c1_im[kc * 9 + (K - 12)];
                }
                ws[OFF_B1 + ((t * 6 + c) * 32 + L) * 2 + j] = v;
            }

    // ---- B2: rbf1[N,32] @ (scale1*rw1) -> 32 real cols (re|im)
    for (int t = 0; t < 2; ++t)
        for (int c = 0; c < 8; ++c)
            for (int j = 0; j < 2; ++j) {
                int K = 4 * c + 2 * half + j;   // rbf index 0..31
                float v = (t == 0) ? s1 * rw1_re[K * HID + r]
                                   : s1 * rw1_im[K * HID + r];
                ws[OFF_B2 + ((t * 8 + c) * 32 + L) * 2 + j] = v;
            }

    // ---- B3: A = [h_re(16) | h_im(16)] (K=32). tile0 = cross2 (16 centers),
    //          tile1 = linear2 packed in cols 0(re),1(im)
    for (int t = 0; t < 2; ++t)
        for (int c = 0; c < 8; ++c)
            for (int j = 0; j < 2; ++j) {
                int K = 4 * c + 2 * half + j;
                float v = 0.f;
                if (t == 0) {
                    v = (K < 16) ? c2_re[r * HID + K] : c2_im[r * HID + (K - 16)];
                } else if (r == 0) {
                    v = (K < 16) ? w2_re[K] : -w2_im[K - 16];
                } else if (r == 1) {
                    v = (K < 16) ? w2_im[K] :  w2_re[K - 16];
                }
                ws[OFF_B3 + ((t * 8 + c) * 32 + L) * 2 + j] = v;
            }

    // ---- B4: rbf2[N,16] @ (scale2*rw2) -> cols 0(re),1(im)
    for (int c = 0; c < 4; ++c)
        for (int j = 0; j < 2; ++j) {
            int K = 4 * c + 2 * half + j;       // rbf2 index 0..15
            float v = 0.f;
            if (r == 0)      v = s2 * rw2_re[K];
            else if (r == 1) v = s2 * rw2_im[K];
            ws[OFF_B4 + (c * 32 + L) * 2 + j] = v;
        }

    // ---- per-lane column constants
    for (int t = 0; t < 2; ++t) {
        int k = r + 16 * t;
        ws[OFF_G1 + t * 32 + L] = fminf(fmaxf(expf(log_g1[k]), 0.01f), 10.f);
        float cs = 0.f;
        for (int d = 0; d < 9; ++d)
            cs += c1_re[k * 9 + d] * c1_re[k * 9 + d] +
                  c1_im[k * 9 + d] * c1_im[k * 9 + d];
        ws[OFF_CS1 + t * 32 + L] = cs;
        ws[OFF_BIA + t * 32 + L] = (t == 0) ? b1_re[r] : b1_im[r];
    }
    ws[OFF_G2 + L] = fminf(fmaxf(expf(log_g2[r]), 0.01f), 10.f);
    {
        float cs = 0.f;
        for (int d = 0; d < HID; ++d)
            cs += c2_re[r * HID + d] * c2_re[r * HID + d] +
                  c2_im[r * HID + d] * c2_im[r * HID + d];
        ws[OFF_CS2 + L] = cs;
    }
    ws[OFF_BOUT + L] = (r == 0) ? b2_re[0] : ((r == 1) ? b2_im[0] : 0.f);
}

// ---------------------------------------------------------------------------
// Main kernel
// ---------------------------------------------------------------------------
__device__ __forceinline__ v8f wmma4(v2f a, v2f b, v8f c) {
    return __builtin_amdgcn_wmma_f32_16x16x4_f32(
        false, a, false, b, (short)0, c, false, false);
}

__device__ __forceinline__ float load_x(const float* __restrict__ xre,
                                        const float* __restrict__ xim,
                                        int row, int K) {
    if (K < 9)                  return xre[row * 9 + K];
    if (K >= 12 && K < 21)      return xim[row * 9 + (K - 12)];
    return 0.f;
}

__global__ void __launch_bounds__(256)
kan_kernel(const float* __restrict__ xre, const float* __restrict__ xim,
           const float* __restrict__ ws, float* __restrict__ out, int nTiles)
{
    __shared__ __align__(16) float lds[8 * 512];   // 2KB per wave
    const int lane = threadIdx.x & 31;
    const int half = lane >> 4, r = lane & 15;
    float* myLds = &lds[(threadIdx.x >> 5) * 512];

    // ---- load all B matrices / constants into registers (persist over loop)
    const v2f* wsB = (const v2f*)ws;
    v2f b1w[4][6], b2w[2][8], b3w[2][8], b4w[4];
#pragma unroll
    for (int t = 0; t < 4; ++t)
#pragma unroll
        for (int c = 0; c < 6; ++c)
            b1w[t][c] = wsB[(OFF_B1 >> 1) + (t * 6 + c) * 32 + lane];
#pragma unroll
    for (int t = 0; t < 2; ++t)
#pragma unroll
        for (int c = 0; c < 8; ++c) {
            b2w[t][c] = wsB[(OFF_B2 >> 1) + (t * 8 + c) * 32 + lane];
            b3w[t][c] = wsB[(OFF_B3 >> 1) + (t * 8 + c) * 32 + lane];
        }
#pragma unroll
    for (int c = 0; c < 4; ++c)
        b4w[c] = wsB[(OFF_B4 >> 1) + c * 32 + lane];

    float g1v[2], cs1v[2], biav[2];
#pragma unroll
    for (int t = 0; t < 2; ++t) {
        g1v[t]  = ws[OFF_G1  + t * 32 + lane];
        cs1v[t] = ws[OFF_CS1 + t * 32 + lane];
        biav[t] = ws[OFF_BIA + t * 32 + lane];
    }
    const float g2v = ws[OFF_G2 + lane];
    const float cs2v = ws[OFF_CS2 + lane];
    const float boutv = ws[OFF_BOUT + lane];

    const int waveId = (blockIdx.x * blockDim.x + threadIdx.x) >> 5;
    const int nWaves = (gridDim.x * blockDim.x) >> 5;

    for (int tile = waveId; tile < nTiles; tile += nWaves) {
        const int baseRow = tile << 4;
        const int nt = tile + nWaves;
        if (nt < nTiles) {  // prefetch next tile's inputs
            __builtin_prefetch(&xre[(nt * 16 + r) * 9], 0, 1);
            __builtin_prefetch(&xim[(nt * 16 + r) * 9], 0, 1);
        }

        // ================= stage 1: linear1 + cross1 (24 WMMAs) ==========
        v8f dl0 = {0,0,0,0,0,0,0,0}, dl1 = {0,0,0,0,0,0,0,0};
        v8f dc0 = {0,0,0,0,0,0,0,0}, dc1 = {0,0,0,0,0,0,0,0};
        float x2p = 0.f;
#pragma unroll
        for (int c = 0; c < 6; ++c) {
            const int K0 = 4 * c + 2 * half;
            float a0 = load_x(xre, xim, baseRow + r, K0);
            float a1 = load_x(xre, xim, baseRow + r, K0 + 1);
            x2p = fmaf(a0, a0, fmaf(a1, a1, x2p));
            v2f a; a.x = a0; a.y = a1;
            dl0 = wmma4(a, b1w[0][c], dl0);
            dl1 = wmma4(a, b1w[1][c], dl1);
            dc0 = wmma4(a, b1w[2][c], dc0);
            dc1 = wmma4(a, b1w[3][c], dc1);
        }
        // |x|^2 per row, broadcast into D-layout v8f
        float x2s = x2p + __shfl_xor(x2p, 16, 32);
        v8f x2v;
#pragma unroll
        for (int i = 0; i < 8; ++i) x2v[i] = __shfl(x2s, 8 * half + i, 32);

        // ================= rbf1 elementwise + LDS transpose ==============
#pragma unroll
        for (int t = 0; t < 2; ++t) {
            v8f dc = t ? dc1 : dc0;
            v8f rb;
#pragma unroll
            for (int i = 0; i < 8; ++i) {
                float dist = x2v[i] + cs1v[t] - 2.f * dc[i];
                rb[i] = fminf(__expf(-g1v[t] * dist), 10.f);
            }
            float* p = &myLds[(r + 16 * t) * 16 + 8 * half];
            ((float4*)p)[0] = make_float4(rb[0], rb[1], rb[2], rb[3]);
            ((float4*)p)[1] = make_float4(rb[4], rb[5], rb[6], rb[7]);
        }
        __asm__ volatile("s_wait_dscnt 0" ::: "memory");

        // ================= stage 2: rbf1 @ rw1, accumulate into h ========
#pragma unroll
        for (int c = 0; c < 8; ++c) {
            const int K0 = 4 * c + 2 * half;
            v2f a; a.x = myLds[K0 * 16 + r]; a.y = myLds[(K0 + 1) * 16 + r];
            dl0 = wmma4(a, b2w[0][c], dl0);
            dl1 = wmma4(a, b2w[1][c], dl1);
        }

        // bias + complex tanh; |h|^2 partial; transpose h into LDS
        v8f h0, h1, h2v;
#pragma unroll
        for (int i = 0; i < 8; ++i) {
            h0[i] = tanhf(dl0[i] + biav[0]);
            h1[i] = tanhf(dl1[i] + biav[1]);
            h2v[i] = h0[i] * h0[i] + h1[i] * h1[i];
        }
#pragma unroll
        for (int m = 1; m < 16; m <<= 1)
#pragma unroll
            for (int i = 0; i < 8; ++i)
                h2v[i] += __shfl_xor(h2v[i], m, 32);
        {
            float* p0 = &myLds[r * 16 + 8 * half];          // k = col (re)
            ((float4*)p0)[0] = make_float4(h0[0], h0[1], h0[2], h0[3]);
            ((float4*)p0)[1] = make_float4(h0[4], h0[5], h0[6], h0[7]);
            float* p1 = &myLds[(r + 16) * 16 + 8 * half];   // k = col+16 (im)
            ((float4*)p1)[0] = make_float4(h1[0], h1[1], h1[2], h1[3]);
            ((float4*)p1)[1] = make_float4(h1[4], h1[5], h1[6], h1[7]);
        }
        __asm__ volatile("s_wait_dscnt 0" ::: "memory");

        // ================= stage 3: cross2 + linear2 (16 WMMAs) ==========
        v8f dcr2 = {0,0,0,0,0,0,0,0};
        v8f dout;
#pragma unroll
        for (int i = 0; i < 8; ++i) dout[i] = boutv;   // b2 bias in cols 0/1
#pragma unroll
        for (int c = 0; c < 8; ++c) {
            const int K0 = 4 * c + 2 * half;
            v2f a; a.x = myLds[K0 * 16 + r]; a.y = myLds[(K0 + 1) * 16 + r];
            dcr2 = wmma4(a, b3w[0][c], dcr2);
            dout = wmma4(a, b3w[1][c], dout);
        }

        // ================= rbf2 + transpose + final accumulate ===========
        v8f r2;
#pragma unroll
        for (int i = 0; i < 8; ++i) {
            float dist = h2v[i] + cs2v - 2.f * dcr2[i];
            r2[i] = fminf(__expf(-g2v * dist), 10.f);
        }
        {
            float* p = &myLds[r * 16 + 8 * half];
            ((float4*)p)[0] = make_float4(r2[0], r2[1], r2[2], r2[3]);
            ((float4*)p)[1] = make_float4(r2[4], r2[5], r2[6], r2[7]);
        }
        __asm__ volatile("s_wait_dscnt 0" ::: "memory");
#pragma unroll
        for (int c = 0; c < 4; ++c) {
            const int K0 = 4 * c + 2 * half;
            v2f a; a.x = myLds[K0 * 16 + r]; a.y = myLds[(K0 + 1) * 16 + r];
            dout = wmma4(a, b4w[c], dout);
        }

        // ================= store interleaved complex output ==============
        if (r < 2) {  // lanes 0,1 -> rows 0..7 ; lanes 16,17 -> rows 8..15
#pragma unroll
            for (int i = 0; i < 8; ++i)
                out[(baseRow + i + 8 * half) * 2 + r] = dout[i];
        }
    }
}

// ---------------------------------------------------------------------------
extern "C" void kernel_launch(void* const* d_in, const int* in_sizes, int n_in,
                              void* d_out, int out_size, void* d_ws, size_t ws_size,
                              hipStream_t stream) {
    const float* xre    = (const float*)d_in[0];
    const float* xim    = (const float*)d_in[1];
    const float* w1_re  = (const float*)d_in[2];
    const float* w1_im  = (const float*)d_in[3];
    const float* b1_re  = (const float*)d_in[4];
    const float* b1_im  = (const float*)d_in[5];
    const float* c1_re  = (const float*)d_in[6];
    const float* c1_im  = (const float*)d_in[7];
    const float* lg1    = (const float*)d_in[8];
    const float* rw1_re = (const float*)d_in[9];
    const float* rw1_im = (const float*)d_in[10];
    const float* w2_re  = (const float*)d_in[11];
    const float* w2_im  = (const float*)d_in[12];
    const float* b2_re  = (const float*)d_in[13];
    const float* b2_im  = (const float*)d_in[14];
    const float* c2_re  = (const float*)d_in[15];
    const float* c2_im  = (const float*)d_in[16];
    const float* lg2    = (const float*)d_in[17];
    const float* rw2_re = (const float*)d_in[18];
    const float* rw2_im = (const float*)d_in[19];
    const float* scale1 = (const float*)d_in[20];
    const float* scale2 = (const float*)d_in[21];

    float* ws  = (float*)d_ws;
    float* out = (float*)d_out;

    const int n = in_sizes[0] / IN_DIM;   // batch size
    const int nTiles = n / 16;

    pack_kernel<<<1, 32, 0, stream>>>(w1_re, w1_im, b1_re, b1_im,
                                      c1_re, c1_im, lg1, rw1_re, rw1_im,
                                      w2_re, w2_im, b2_re, b2_im,
                                      c2_re, c2_im, lg2, rw2_re, rw2_im,
                                      scale1, scale2, ws);

    kan_kernel<<<2048, 256, 0, stream>>>(xre, xim, ws, out, nTiles);
}